// MaskLayer_45999099740244
// MI455X (gfx1250) — compile-verified
//
#include <hip/hip_runtime.h>
#include <stdint.h>

// ---------------------------------------------------------------------------
// MaskLayer for MI455X (gfx1250): bandwidth-bound (384 MiB logical traffic,
// ~0.17 GFLOP). 192MB L2 keeps the 128MiB input resident between pass 1 and
// pass 3 -> ~256 MiB HBM ~= 11us at 23.3 TB/s. No matmul -> WMMA N/A; the
// CDNA5-specific paths used are TDM tensor_load_to_lds + s_wait_tensorcnt
// (broadcast of the 16KB mask table into LDS) and global_prefetch streaming.
// mask_apply is defined FIRST so the disasm snippet shows the TDM lowering.
// ---------------------------------------------------------------------------

#define N_SP        128
#define DCH         2048
#define NPOS        (N_SP * N_SP)     // 16384 spatial positions
#define SEG         128               // argmax segments (blocks in pass 1)
#define POS_PER_SEG (NPOS / SEG)      // 128
#define POS_PER_BLK 16                // positions per block in pass 3
#define THREADS     512               // 4 channels per thread (2048/4)

typedef unsigned int u32;
typedef __attribute__((ext_vector_type(4))) u32   u32x4;
typedef __attribute__((ext_vector_type(8))) int   i32x8;
typedef __attribute__((ext_vector_type(4))) int   i32x4;
typedef __attribute__((ext_vector_type(4))) float f32x4;

#if defined(__HIP_DEVICE_COMPILE__) && \
    __has_builtin(__builtin_amdgcn_tensor_load_to_lds) && \
    __has_builtin(__builtin_amdgcn_s_wait_tensorcnt)
#define MASK_USE_TDM 1
#else
#define MASK_USE_TDM 0
#endif

#if MASK_USE_TDM
__device__ __forceinline__ u32 lds_byte_offset(void* p) {
  // generic -> LDS addrspace cast; AS(3) pointers are byte offsets from LDS base
  return (u32)(unsigned long long)(__attribute__((address_space(3))) char*)p;
}
#endif

// ---------------------------------------------------------------------------
// Pass 3: out = x * max(-1, 1 - (|i-i_max|+|j-j_max|)/128).
// The 16KB (i_max|j_max) table is DMA'd into LDS once per block by the TDM
// (wave 0 issues tensor_load_to_lds, waits on TENSORcnt, then barrier).
// x re-read hits L2 (192MB); stores are non-temporal to keep x resident.
// ---------------------------------------------------------------------------
__global__ __launch_bounds__(THREADS) void mask_apply(
    const float* __restrict__ x, const float* __restrict__ tbl,
    float* __restrict__ out) {
  __shared__ float sh[2 * DCH];  // 16 KB

#if MASK_USE_TDM
  if (threadIdx.x < 32) {        // one wave issues the TDM op (EXEC ignored by TDM)
    const unsigned long long ga = (unsigned long long)(const void*)tbl;
    u32x4 g0;
    g0.x = 1u;                                             // count=1, user mode
    g0.y = lds_byte_offset((void*)sh);                     // lds_addr (bytes)
    g0.z = (u32)ga;                                        // global_addr[31:0]
    g0.w = (u32)((ga >> 32) & 0x01FFFFFFull) | (2u << 30); // addr[56:32] | type=2
    i32x8 g1;
    g1[0] = 0x00020000;                                 // data_size=2 (4 bytes)
    g1[1] = (int)(((u32)(2 * DCH) & 0xFFFFu) << 16);    // tensor_dim0[15:0]=4096
    g1[2] = (int)(1u << 16);                            // dim0 hi=0 ; tensor_dim1=1
    g1[3] = (int)((u32)(2 * DCH) << 16);                // dim1 hi=0 ; tile_dim0=4096
    g1[4] = 1;                                          // tile_dim1=1, tile_dim2=0
    g1[5] = 2 * DCH;                                    // tensor_dim0_stride=4096
    g1[6] = 0;
    g1[7] = 0;
    i32x4 gz = {0, 0, 0, 0};
#if defined(__clang_major__) && (__clang_major__ >= 23)
    i32x8 gz8 = {0, 0, 0, 0, 0, 0, 0, 0};
    __builtin_amdgcn_tensor_load_to_lds(g0, g1, gz, gz, gz8, 0);
#else
    __builtin_amdgcn_tensor_load_to_lds(g0, g1, gz, gz, 0);
#endif
    __builtin_amdgcn_s_wait_tensorcnt((short)0);
  }
  __syncthreads();
#else
  for (int t = threadIdx.x; t < 2 * DCH; t += THREADS) sh[t] = tbl[t];
  __syncthreads();
#endif

  const int tid = threadIdx.x;
  const int c4  = tid << 2;
  const f32x4 im = *reinterpret_cast<const f32x4*>(&sh[c4]);
  const f32x4 jm = *reinterpret_cast<const f32x4*>(&sh[DCH + c4]);
  const int   pbeg  = blockIdx.x * POS_PER_BLK;
  const float inv_n = 1.0f / (float)N_SP;
  const float* __restrict__ xin  = x   + (size_t)pbeg * DCH + c4;
  float* __restrict__       oout = out + (size_t)pbeg * DCH + c4;

#pragma unroll
  for (int k = 0; k < POS_PER_BLK; ++k) {
    const int p = pbeg + k;
    const float fi = (float)(p >> 7);            // i = p / n
    const float fj = (float)(p & (N_SP - 1));    // j = p % n
    const f32x4 v = *reinterpret_cast<const f32x4*>(xin + (size_t)k * DCH);
    f32x4 r;
    r.x = v.x * fmaxf(-1.0f, 1.0f - (fabsf(fi - im.x) + fabsf(fj - jm.x)) * inv_n);
    r.y = v.y * fmaxf(-1.0f, 1.0f - (fabsf(fi - im.y) + fabsf(fj - jm.y)) * inv_n);
    r.z = v.z * fmaxf(-1.0f, 1.0f - (fabsf(fi - im.z) + fabsf(fj - jm.z)) * inv_n);
    r.w = v.w * fmaxf(-1.0f, 1.0f - (fabsf(fi - im.w) + fabsf(fj - jm.w)) * inv_n);
    __builtin_nontemporal_store(r, reinterpret_cast<f32x4*>(oout + (size_t)k * DCH));
  }
}

// ---------------------------------------------------------------------------
// Pass 1: per-(segment, channel) partial argmax.  x[p*DCH + c], channels are
// the unit-stride axis -> thread t owns channels 4t..4t+3 via 16B loads; each
// wave reads 512B contiguous per iteration. Strict '>' keeps first occurrence.
// Chunked 4x loop: one pointer increment + one clamped prefetch per 32KB row
// chunk; the 4 loads use constant immediate offsets (0/8K/16K/24K).
// ---------------------------------------------------------------------------
__global__ __launch_bounds__(THREADS) void mask_argmax_partial(
    const float* __restrict__ x, float* __restrict__ pval, int* __restrict__ pidx) {
  const int tid  = threadIdx.x;
  const int c4   = tid << 2;
  const int pbeg = blockIdx.x * POS_PER_SEG;
  const float* ptr = x + (size_t)pbeg * DCH + c4;

  f32x4 bv = {-__builtin_inff(), -__builtin_inff(), -__builtin_inff(), -__builtin_inff()};
  i32x4 bp = {pbeg, pbeg, pbeg, pbeg};

#pragma unroll 1
  for (int k = 0; k < POS_PER_SEG; k += 4) {
    // stream-prefetch 64KB ahead within this segment (-> global_prefetch_b8)
    const float* pf = ptr + ((k + 8 < POS_PER_SEG) ? (size_t)(8 * DCH) : 0u);
    __builtin_prefetch(pf, 0, 1);
#pragma unroll
    for (int u = 0; u < 4; ++u) {
      const f32x4 v = *reinterpret_cast<const f32x4*>(ptr + (size_t)u * DCH);
      const int p = pbeg + k + u;
      if (v.x > bv.x) { bv.x = v.x; bp.x = p; }
      if (v.y > bv.y) { bv.y = v.y; bp.y = p; }
      if (v.z > bv.z) { bv.z = v.z; bp.z = p; }
      if (v.w > bv.w) { bv.w = v.w; bp.w = p; }
    }
    ptr += 4 * DCH;
  }
  *reinterpret_cast<f32x4*>(pval + (size_t)blockIdx.x * DCH + c4) = bv;
  *reinterpret_cast<i32x4*>(pidx + (size_t)blockIdx.x * DCH + c4) = bp;
}

// ---------------------------------------------------------------------------
// Pass 2: reduce SEG partials per channel (ascending segment scan + strict '>'
// == global first-occurrence argmax), emit reference's transposed coords:
// i_max = idx % n, j_max = idx / n, as a 4096-float table [i_max | j_max].
// ---------------------------------------------------------------------------
__global__ void mask_argmax_final(const float* __restrict__ pval,
                                  const int* __restrict__ pidx,
                                  float* __restrict__ tbl) {
  const int c = blockIdx.x * blockDim.x + threadIdx.x;
  if (c >= DCH) return;
  float bv = -__builtin_inff();
  int   bp = 0;
  for (int s = 0; s < SEG; ++s) {
    const float v = pval[(size_t)s * DCH + c];
    if (v > bv) { bv = v; bp = pidx[(size_t)s * DCH + c]; }
  }
  tbl[c]       = (float)(bp % N_SP);   // i_max (reference uses idx % n for rows)
  tbl[DCH + c] = (float)(bp / N_SP);   // j_max
}

// ---------------------------------------------------------------------------
// Host launcher. Workspace layout:
//   [0, 1MiB)        float pval[SEG][DCH]
//   [1MiB, 2MiB)     int   pidx[SEG][DCH]
//   [2MiB, 2MiB+16K) float tbl[2*DCH]   ({i_max | j_max})
// ---------------------------------------------------------------------------
extern "C" void kernel_launch(void* const* d_in, const int* in_sizes, int n_in,
                              void* d_out, int out_size, void* d_ws, size_t ws_size,
                              hipStream_t stream) {
  (void)in_sizes; (void)n_in; (void)out_size; (void)ws_size;
  const float* x = (const float*)d_in[0];
  float* out  = (float*)d_out;
  float* pval = (float*)d_ws;
  int*   pidx = (int*)((char*)d_ws + (size_t)SEG * DCH * sizeof(float));
  float* tbl  = (float*)((char*)d_ws + (size_t)SEG * DCH * (sizeof(float) + sizeof(int)));

  mask_argmax_partial<<<SEG, THREADS, 0, stream>>>(x, pval, pidx);
  mask_argmax_final<<<DCH / 256, 256, 0, stream>>>(pval, pidx, tbl);
  mask_apply<<<NPOS / POS_PER_BLK, THREADS, 0, stream>>>(x, tbl, out);
}